// GraphCare_32117765439835
// MI455X (gfx1250) — compile-verified
//
#include <hip/hip_runtime.h>
#include <hip/hip_bf16.h>
#include <math.h>

// Problem constants (reference: B,V,NV,NR,N,E,D,H,OUT,L = 4,20,8000,200,8000,200000,128,128,64,3)
#define GC_B   4
#define GC_V   20
#define GC_NV  8000
#define GC_NR  200
#define GC_N   8000
#define GC_E   200000
#define GC_D   128
#define GC_H   128
#define GC_OUT 64
#define GC_L   3
#define GC_DECAY 0.03f

typedef float v2f __attribute__((ext_vector_type(2)));
typedef float v8f __attribute__((ext_vector_type(8)));

// ---------------------------------------------------------------- utilities

__global__ void zero_f_kernel(float* p, int n) {
    int i = blockIdx.x * blockDim.x + threadIdx.x;
    if (i < n) p[i] = 0.0f;
}

__global__ void counts_kernel(const int* batch, int* counts, int n) {
    int i = blockIdx.x * blockDim.x + threadIdx.x;
    if (i < n) atomicAdd(&counts[batch[i]], 1);
}

__global__ void starts_kernel(const int* counts, int* starts) {
    if (blockIdx.x == 0 && threadIdx.x == 0) {
        int s = 0;
        for (int b = 0; b < GC_B; ++b) { starts[b] = s; s += counts[b]; }
    }
}

// rel_proj[r,h] = rel_emb[r,:] @ lin_w[:,h] + lin_b[h]    (200 x 128, tiny)
__global__ void rel_proj_kernel(const float* __restrict__ rel_emb,
                                const float* __restrict__ lin_w,
                                const float* __restrict__ lin_b,
                                float* __restrict__ rel_proj) {
    __shared__ float row[GC_D];
    int r = blockIdx.x, h = threadIdx.x;
    row[h] = rel_emb[r * GC_D + h];
    __syncthreads();
    float acc = lin_b[h];
#pragma unroll 8
    for (int d = 0; d < GC_D; ++d) acc += row[d] * lin_w[d * GC_H + h];
    rel_proj[r * GC_H + h] = acc;
}

// ------------------------------------------------------- WMMA fp32 GEMM
// out[M=grid*16, 128] = act( (A0[gidx] (+A1)) @ Bm[128,128] + bias )
// One block = 256 threads = 8 waves; block owns one 16-row M tile,
// wave w owns columns [16w, 16w+16). K-loop: 32x V_WMMA_F32_16X16X4_F32.
__global__ void gemm_wmma_kernel(const float* __restrict__ A0,
                                 const int*  __restrict__ gidx,   // nullable row gather for A0
                                 const float* __restrict__ A1,    // nullable, added (indexed by out row)
                                 const float* __restrict__ Bm,    // [128,128] row major
                                 const float* __restrict__ bias,  // [128]
                                 float* __restrict__ out,
                                 int relu) {
    __shared__ float As[16 * 132];   // stride 132: bank-conflict free, float2-aligned
    const int t = threadIdx.x;
    const int mbase = blockIdx.x * 16;

    // cooperative A-tile load: 16 rows x 128 cols as float4
    for (int it = t; it < 512; it += 256) {
        int r  = it >> 5;        // 0..15
        int c4 = it & 31;        // 0..31 float4 within row
        int row = mbase + r;
        int srow = gidx ? gidx[row] : row;
        float4 v = ((const float4*)A0)[srow * 32 + c4];
        if (A1) {
            float4 w = ((const float4*)A1)[row * 32 + c4];
            v.x += w.x; v.y += w.y; v.z += w.z; v.w += w.w;
        }
        *(float4*)&As[r * 132 + c4 * 4] = v;
    }
    __syncthreads();

    const int wave = t >> 5;
    const int lane = t & 31;
    const int m    = lane & 15;           // A row within tile / C column
    const int koff = (lane >> 4) << 1;    // 0 or 2 (K sub-offset per half-wave)
    const int hbase = wave * 16;
    const int ncol  = hbase + (lane & 15);

    v8f c = {0.f, 0.f, 0.f, 0.f, 0.f, 0.f, 0.f, 0.f};
#pragma unroll
    for (int k0 = 0; k0 < 128; k0 += 4) {
        float2 av = *(const float2*)&As[m * 132 + k0 + koff];
        v2f a; a[0] = av.x; a[1] = av.y;
        v2f b;
        b[0] = Bm[(k0 + koff)     * GC_H + ncol];
        b[1] = Bm[(k0 + koff + 1) * GC_H + ncol];
        c = __builtin_amdgcn_wmma_f32_16x16x4_f32(false, a, false, b, (short)0, c,
                                                  false, false);
    }

    // C/D layout: vgpr g -> row g + 8*(lane>=16), col lane%16
    const int col   = lane & 15;
    const int rbase = (lane >> 4) * 8;
    const float bv  = bias[hbase + col];
#pragma unroll
    for (int g = 0; g < 8; ++g) {
        float v = c[g] + bv;
        if (relu) v = fmaxf(v, 0.0f);
        out[(mbase + rbase + g) * GC_H + hbase + col] = v;
    }
}

// out[i] = rows[i,:128] . w[:128]  (+ *bias)   — one wave per row
__global__ void row_dot_kernel(const float* __restrict__ rows,
                               const float* __restrict__ w,
                               const float* __restrict__ bias, // nullable scalar
                               float* __restrict__ out, int M) {
    int i = (blockIdx.x * blockDim.x + threadIdx.x) >> 5;
    if (i >= M) return;
    int lane = threadIdx.x & 31;
    float4 xv = ((const float4*)(rows + i * 128))[lane];
    float4 wv = ((const float4*)w)[lane];
    float p = xv.x * wv.x + xv.y * wv.y + xv.z * wv.z + xv.w * wv.w;
#pragma unroll
    for (int off = 16; off; off >>= 1) p += __shfl_xor(p, off, 32);
    if (lane == 0) out[i] = p + (bias ? bias[0] : 0.0f);
}

// beta0[b] = tanh(visit_node[b,0,:] . beta_w_l + beta_b_l) * exp(DECAY*V)
__global__ void beta_kernel(const float* __restrict__ visit_node,
                            const float* __restrict__ beta_w_l,
                            const float* __restrict__ beta_b_l,
                            float* __restrict__ beta0) {
    __shared__ float red[256];
    int b = blockIdx.x, t = threadIdx.x;
    float acc = 0.0f;
    for (int n = t; n < GC_NV; n += 256)
        acc += visit_node[(size_t)b * GC_V * GC_NV + n] * beta_w_l[n];
    red[t] = acc; __syncthreads();
    for (int s = 128; s; s >>= 1) { if (t < s) red[t] += red[t + s]; __syncthreads(); }
    if (t == 0) beta0[b] = tanhf(red[0] + beta_b_l[0]) * expf(GC_DECAY * (float)GC_V);
}

// attn0[b,n] = softmax_v(visit_node[b,v,n]*s_pad + alpha_b)[v=0] * beta0[b]
__global__ void attn_kernel(const float* __restrict__ visit_node,
                            const float* __restrict__ sdot,
                            const int*  __restrict__ starts,
                            const int*  __restrict__ counts,
                            const float* __restrict__ beta0,
                            const float* __restrict__ alpha_b_l,
                            float* __restrict__ attn0) {
    int idx = blockIdx.x * blockDim.x + threadIdx.x;
    if (idx >= GC_B * GC_NV) return;
    int b = idx / GC_NV, n = idx - b * GC_NV;
    float s = 0.0f;
    if (n < counts[b]) s = sdot[starts[b] + n];
    float ab = alpha_b_l[0];
    float lg[GC_V];
    float mx = -1e30f;
#pragma unroll
    for (int v = 0; v < GC_V; ++v) {
        float lv = visit_node[((size_t)b * GC_V + v) * GC_NV + n] * s + ab;
        lg[v] = lv;
        mx = fmaxf(mx, lv);
    }
    float denom = 0.0f, num = 0.0f;
#pragma unroll
    for (int v = 0; v < GC_V; ++v) {
        float e = expf(lg[v] - mx);
        denom += e;
        if (v == 0) num = e;
    }
    attn0[idx] = (num / denom) * beta0[b];
}

// per edge (one wave): msg = relu(x[src]*attn + wr[rel]*rel_proj[rel]); agg[dst] += msg
__global__ void edge_kernel(const int* __restrict__ e_src, const int* __restrict__ e_dst,
                            const int* __restrict__ rel_ids, const int* __restrict__ batch,
                            const int* __restrict__ node_ids,
                            const float* __restrict__ attn0, const float* __restrict__ wr,
                            const float* __restrict__ rel_proj,
                            const float* __restrict__ x, float* __restrict__ agg) {
    int e = (blockIdx.x * blockDim.x + threadIdx.x) >> 5;
    if (e >= GC_E) return;
    int lane = threadIdx.x & 31;
    int s = e_src[e], d = e_dst[e], r = rel_ids[e];
    float attn = attn0[batch[s] * GC_NV + node_ids[s]];
    float wrv  = wr[r];
    float4 xv = ((const float4*)(x + (size_t)s * GC_H))[lane];
    float4 rv = ((const float4*)(rel_proj + (size_t)r * GC_H))[lane];
    float* ap = agg + (size_t)d * GC_H + lane * 4;
    atomicAdd(ap + 0, fmaxf(xv.x * attn + wrv * rv.x, 0.0f));
    atomicAdd(ap + 1, fmaxf(xv.y * attn + wrv * rv.y, 0.0f));
    atomicAdd(ap + 2, fmaxf(xv.z * attn + wrv * rv.z, 0.0f));
    atomicAdd(ap + 3, fmaxf(xv.w * attn + wrv * rv.w, 0.0f));
}

// per-graph mean pool of x  +  EHR-weighted mean of node embeddings
__global__ void pool_kernel(const float* __restrict__ x,
                            const int* __restrict__ starts, const int* __restrict__ counts,
                            const float* __restrict__ ehr, const float* __restrict__ emb,
                            float* __restrict__ xg, float* __restrict__ xnr) {
    int b = blockIdx.x, t = threadIdx.x;  // blockDim = 128
    int st = starts[b], cn = counts[b];
    float acc = 0.0f;
    for (int i = 0; i < cn; ++i) acc += x[(size_t)(st + i) * GC_H + t];
    xg[b * GC_H + t] = acc / (float)cn;

    __shared__ float red[128];
    float es = 0.0f;
    for (int n = t; n < GC_NV; n += 128) es += ehr[(size_t)b * GC_NV + n];
    red[t] = es; __syncthreads();
    for (int s2 = 64; s2; s2 >>= 1) { if (t < s2) red[t] += red[t + s2]; __syncthreads(); }
    float esum = red[0];

    float acc2 = 0.0f;
    for (int n = 0; n < GC_NV; ++n)
        acc2 += ehr[(size_t)b * GC_NV + n] * emb[(size_t)n * GC_D + t];
    xnr[b * GC_D + t] = acc2 / esum;
}

// xnp[b,h] = xnr[b,:] @ lin_w[:,h] + lin_b[h]
__global__ void xnode_proj_kernel(const float* __restrict__ xnr,
                                  const float* __restrict__ lin_w,
                                  const float* __restrict__ lin_b,
                                  float* __restrict__ xnp) {
    __shared__ float row[GC_D];
    int b = blockIdx.x, h = threadIdx.x;
    row[h] = xnr[b * GC_D + h];
    __syncthreads();
    float acc = lin_b[h];
#pragma unroll 8
    for (int d = 0; d < GC_D; ++d) acc += row[d] * lin_w[d * GC_H + h];
    xnp[b * GC_H + h] = acc;
}

// logits[b,o] = concat(xg, xnp)[b,:] @ mlp_w + mlp_b
__global__ void mlp_kernel(const float* __restrict__ xg, const float* __restrict__ xnp,
                           const float* __restrict__ mlp_w, const float* __restrict__ mlp_b,
                           float* __restrict__ out) {
    int t = threadIdx.x;             // 256 = B*OUT
    int b = t >> 6, o = t & 63;
    float acc = mlp_b[o];
#pragma unroll 8
    for (int j = 0; j < GC_H; ++j) acc += xg[b * GC_H + j] * mlp_w[j * GC_OUT + o];
#pragma unroll 8
    for (int j = 0; j < GC_H; ++j) acc += xnp[b * GC_H + j] * mlp_w[(GC_H + j) * GC_OUT + o];
    out[t] = acc;
}

// ---------------------------------------------------------------- launch

extern "C" void kernel_launch(void* const* d_in, const int* in_sizes, int n_in,
                              void* d_out, int out_size, void* d_ws, size_t ws_size,
                              hipStream_t stream) {
    (void)in_sizes; (void)n_in; (void)out_size; (void)ws_size;

    const int*   node_ids   = (const int*)d_in[0];
    const int*   rel_ids    = (const int*)d_in[1];
    const int*   e_src      = (const int*)d_in[2];
    const int*   e_dst      = e_src + GC_E;
    const int*   batch      = (const int*)d_in[3];
    const float* visit_node = (const float*)d_in[4];
    const float* ehr        = (const float*)d_in[5];
    const float* node_emb   = (const float*)d_in[6];
    const float* rel_emb    = (const float*)d_in[7];
    const float* lin_w      = (const float*)d_in[8];
    const float* lin_b      = (const float*)d_in[9];
    const float* alpha_w    = (const float*)d_in[10];
    const float* alpha_b    = (const float*)d_in[11];
    const float* beta_w     = (const float*)d_in[12];
    const float* beta_b     = (const float*)d_in[13];
    const float* conv_w     = (const float*)d_in[14];
    const float* conv_b     = (const float*)d_in[15];
    const float* wr_w       = (const float*)d_in[16];
    const float* wr_b       = (const float*)d_in[17];
    const float* mlp_w      = (const float*)d_in[18];
    const float* mlp_b      = (const float*)d_in[19];
    float* out = (float*)d_out;

    size_t off = 0;
    auto alloc = [&](size_t bytes) -> char* {
        char* p = (char*)d_ws + off;
        off = (off + bytes + 255) & ~(size_t)255;
        return p;
    };
    float* xa       = (float*)alloc((size_t)GC_N * GC_H * 4);
    float* xb       = (float*)alloc((size_t)GC_N * GC_H * 4);
    float* agg      = (float*)alloc((size_t)GC_N * GC_H * 4);
    float* rel_proj = (float*)alloc((size_t)GC_NR * GC_H * 4);
    float* sdot     = (float*)alloc((size_t)GC_N * 4);
    float* attn0    = (float*)alloc((size_t)GC_B * GC_NV * 4);
    float* beta0    = (float*)alloc((size_t)GC_B * 4);
    float* wr       = (float*)alloc((size_t)GC_NR * 4);
    int*   counts   = (int*)alloc((size_t)GC_B * 4);
    int*   starts   = (int*)alloc((size_t)GC_B * 4);
    float* xg       = (float*)alloc((size_t)GC_B * GC_H * 4);
    float* xnr      = (float*)alloc((size_t)GC_B * GC_D * 4);
    float* xnp      = (float*)alloc((size_t)GC_B * GC_H * 4);

    // graph bookkeeping
    zero_f_kernel<<<1, 256, 0, stream>>>((float*)counts, GC_B);
    counts_kernel<<<(GC_N + 255) / 256, 256, 0, stream>>>(batch, counts, GC_N);
    starts_kernel<<<1, 1, 0, stream>>>(counts, starts);

    // relation projection table (collapses the [E,128]@[128,128] GEMM to 200 rows)
    rel_proj_kernel<<<GC_NR, GC_H, 0, stream>>>(rel_emb, lin_w, lin_b, rel_proj);

    // x = emb[node_ids] @ lin_w + lin_b     (WMMA, gathered A)
    gemm_wmma_kernel<<<GC_N / 16, 256, 0, stream>>>(node_emb, node_ids, nullptr,
                                                    lin_w, lin_b, xa, 0);

    float* xin = xa;
    float* xout = xb;
    for (int l = 0; l < GC_L; ++l) {
        row_dot_kernel<<<(GC_N * 32) / 256, 256, 0, stream>>>(
            xin, alpha_w + (size_t)l * GC_H, nullptr, sdot, GC_N);
        row_dot_kernel<<<(GC_NR * 32 + 255) / 256, 256, 0, stream>>>(
            rel_proj, wr_w + (size_t)l * GC_H, wr_b + l, wr, GC_NR);
        beta_kernel<<<GC_B, 256, 0, stream>>>(
            visit_node, beta_w + (size_t)l * GC_NV, beta_b + l, beta0);
        attn_kernel<<<(GC_B * GC_NV + 255) / 256, 256, 0, stream>>>(
            visit_node, sdot, starts, counts, beta0, alpha_b + l, attn0);
        zero_f_kernel<<<(GC_N * GC_H + 255) / 256, 256, 0, stream>>>(agg, GC_N * GC_H);
        edge_kernel<<<(GC_E * 32 + 255) / 256, 256, 0, stream>>>(
            e_src, e_dst, rel_ids, batch, node_ids, attn0, wr, rel_proj, xin, agg);
        // x = relu((agg + x) @ conv_w[l] + conv_b[l])    (WMMA, fused add+bias+relu)
        gemm_wmma_kernel<<<GC_N / 16, 256, 0, stream>>>(
            agg, nullptr, xin, conv_w + (size_t)l * GC_H * GC_H,
            conv_b + (size_t)l * GC_H, xout, 1);
        float* tmp = xin; xin = xout; xout = tmp;
    }

    pool_kernel<<<GC_B, GC_H, 0, stream>>>(xin, starts, counts, ehr, node_emb, xg, xnr);
    xnode_proj_kernel<<<GC_B, GC_H, 0, stream>>>(xnr, lin_w, lin_b, xnp);
    mlp_kernel<<<1, GC_B * GC_OUT, 0, stream>>>(xg, xnp, mlp_w, mlp_b, out);
}